// GraphEncoder_46660524704196
// MI455X (gfx1250) — compile-verified
//
#include <hip/hip_runtime.h>
#include <math.h>

// ---------------- problem constants (match reference) ----------------
#define NB     32
#define NPER   2048
#define NTOT   65536           // NB * NPER
#define EDG    1048576
#define DF     128
#define NCLS   8
#define BN_EPS 1e-5f

// gemm epilogue flags
#define F_RELU_IN   1
#define F_RELU_OUT  2
#define F_BIAS      4
#define F_GBIAS     8

typedef float v2f __attribute__((ext_vector_type(2)));
typedef float v8f __attribute__((ext_vector_type(8)));

// =====================================================================
// WMMA GEMM:  Out[NTOT x 128] = op(A)[NTOT x 128] @ W[128 x 128] (+bias)
// block = 256 threads = 8 waves; block computes a 16x128 output strip.
// A tile (16x128 fp32, 8KB) staged in LDS once; each wave runs 32 steps
// of v_wmma_f32_16x16x4_f32 (full fp32 precision).
// =====================================================================
__global__ __launch_bounds__(256) void gemm_k128_wmma(
    const float* __restrict__ A, const float* __restrict__ W,
    const float* __restrict__ bias, const float* __restrict__ gbias,
    float* __restrict__ Out, int flags) {
  __shared__ float As[16 * 128];

  const int m0 = blockIdx.x * 16;

  // cooperative A-tile load (16 rows x 128 cols), optional input relu
  {
    const float4* src = (const float4*)(A + (size_t)m0 * DF);
    float4* dst = (float4*)As;
    for (int i = threadIdx.x; i < 16 * DF / 4; i += 256) {
      float4 v = src[i];
      if (flags & F_RELU_IN) {
        v.x = fmaxf(v.x, 0.f); v.y = fmaxf(v.y, 0.f);
        v.z = fmaxf(v.z, 0.f); v.w = fmaxf(v.w, 0.f);
      }
      dst[i] = v;
    }
  }
  __syncthreads();

  const int wv   = threadIdx.x >> 5;   // wave 0..7 -> column strip
  const int n0   = wv * 16;
  const int lane = threadIdx.x & 31;
  const int half = lane >> 4;          // 0: lanes 0-15, 1: lanes 16-31
  const int lm   = lane & 15;          // row (A) / col (B,C) within tile

  v8f acc = {};
  const float* Wp = W + n0 + lm;       // W[k][n0+lm], row-major

  #pragma unroll
  for (int kk = 0; kk < 32; ++kk) {
    const int k0 = kk * 4;
    const int ka = k0 + half * 2;
    // A 16x4 frag: lane<16 holds K={k0,k0+1}, lane>=16 holds K={k0+2,k0+3}
    v2f a;
    a.x = As[lm * DF + ka];
    a.y = As[lm * DF + ka + 1];
    // B 4x16 frag: vgpr0 = row ka, vgpr1 = row ka+1, column n0+lm
    v2f b;
    b.x = Wp[(size_t)ka * DF];
    b.y = Wp[(size_t)(ka + 1) * DF];
    acc = __builtin_amdgcn_wmma_f32_16x16x4_f32(
        false, a, false, b, (short)0, acc, false, false);
  }

  // epilogue: C/D layout -> vgpr r == rows m0+r (lanes 0-15) / m0+r+8
  const int col = n0 + lm;
  float badd = 0.f;
  if (flags & F_BIAS)  badd = bias[col];
  if (flags & F_GBIAS) badd += gbias[((m0 >> 11) << 7) + col];  // graph = row>>11
  #pragma unroll
  for (int r = 0; r < 8; ++r) {
    const int row = m0 + r + half * 8;
    float v = acc[r] + badd;
    if (flags & F_RELU_OUT) v = fmaxf(v, 0.f);
    Out[(size_t)row * DF + col] = v;
  }
}

// =====================================================================
// degree / norm
// =====================================================================
__global__ void deg_init(float* __restrict__ dinv) {
  int i = blockIdx.x * blockDim.x + threadIdx.x;
  if (i < NTOT) dinv[i] = 1.0f;               // self loop
}
__global__ void deg_count(const int* __restrict__ dst, float* __restrict__ dinv) {
  int e = blockIdx.x * blockDim.x + threadIdx.x;
  if (e < EDG) unsafeAtomicAdd(&dinv[dst[e]], 1.0f);
}
__global__ void deg_finish(float* __restrict__ dinv) {
  int i = blockIdx.x * blockDim.x + threadIdx.x;
  if (i < NTOT) dinv[i] = rsqrtf(dinv[i]);
}

// agg[i][:] = dinv[i]^2 * t[i][:] + b[:]   (self loop + bias fused)
__global__ void agg_init(const float* __restrict__ t, const float* __restrict__ dinv,
                         const float* __restrict__ bias, float* __restrict__ agg) {
  int i4 = blockIdx.x * blockDim.x + threadIdx.x;     // float4 index
  if (i4 >= NTOT * DF / 4) return;
  int row = i4 >> 5;
  int cg  = i4 & 31;
  float d = dinv[row];
  float s = d * d;
  float4 tv = ((const float4*)t)[i4];
  float4 bv = ((const float4*)bias)[cg];
  float4 o;
  o.x = s * tv.x + bv.x; o.y = s * tv.y + bv.y;
  o.z = s * tv.z + bv.z; o.w = s * tv.w + bv.w;
  ((float4*)agg)[i4] = o;
}

// one wave per edge: lane handles 4 features; f32 atomics land in L2
__global__ __launch_bounds__(256) void edge_scatter(
    const float* __restrict__ t, const int* __restrict__ src,
    const int* __restrict__ dst, const float* __restrict__ dinv,
    float* __restrict__ agg) {
  int wave = blockIdx.x * 8 + (threadIdx.x >> 5);
  int lane = threadIdx.x & 31;
  if (wave >= EDG) return;
  int s = src[wave], d = dst[wave];
  float nrm = dinv[s] * dinv[d];
  float4 v = ((const float4*)(t + (size_t)s * DF))[lane];
  float* o = agg + (size_t)d * DF + lane * 4;
  unsafeAtomicAdd(o + 0, nrm * v.x);
  unsafeAtomicAdd(o + 1, nrm * v.y);
  unsafeAtomicAdd(o + 2, nrm * v.z);
  unsafeAtomicAdd(o + 3, nrm * v.w);
}

// =====================================================================
// per-graph bias: rows 128..383 of ff_w applied to the (constant per
// graph) src/dest vectors, plus ff_b.   gbias[32][128]
// =====================================================================
__global__ void graph_bias(const float* __restrict__ h2,
                           const int* __restrict__ sidx, const int* __restrict__ didx,
                           const float* __restrict__ ffw, const float* __restrict__ ffb,
                           float* __restrict__ gbias) {
  int b = blockIdx.x;        // 0..31
  int n = threadIdx.x;       // 0..127
  const float* hs = h2 + (size_t)(sidx[b] + b * NPER) * DF;
  const float* hd = h2 + (size_t)(didx[b] + b * NPER) * DF;
  float acc = ffb[n];
  for (int k = 0; k < DF; ++k) {
    acc += hs[k] * ffw[(size_t)(DF + k) * DF + n];
    acc += hd[k] * ffw[(size_t)(2 * DF + k) * DF + n];
  }
  gbias[b * DF + n] = acc;
}

// =====================================================================
// batchnorm over nodes
// =====================================================================
__global__ void stats_zero(float* __restrict__ stats) {
  if (threadIdx.x < 2 * DF) stats[threadIdx.x] = 0.f;
}
__global__ __launch_bounds__(128) void bn_reduce(const float* __restrict__ h,
                                                 float* __restrict__ stats) {
  int c  = threadIdx.x;                 // column
  int r0 = blockIdx.x * 128;            // 128 rows per block
  float s = 0.f, q = 0.f;
  for (int r = 0; r < 128; ++r) {
    float v = h[(size_t)(r0 + r) * DF + c];
    s += v; q += v * v;
  }
  unsafeAtomicAdd(&stats[c], s);
  unsafeAtomicAdd(&stats[DF + c], q);
}
__global__ void bn_apply(const float* __restrict__ h, const float* __restrict__ stats,
                         const float* __restrict__ gamma, const float* __restrict__ beta,
                         float* __restrict__ out) {
  size_t i = (size_t)blockIdx.x * blockDim.x + threadIdx.x;
  if (i >= (size_t)NTOT * DF) return;
  int c = (int)(i & (DF - 1));
  const float invN = 1.0f / (float)NTOT;
  float mu  = stats[c] * invN;
  float var = stats[DF + c] * invN - mu * mu;
  float sc  = rsqrtf(var + BN_EPS) * gamma[c];
  float v   = (h[i] - mu) * sc + beta[c];
  out[i] = fmaxf(v, 0.f);
}

// =====================================================================
// head: [N,128] @ [128,8] + b  ->  log_softmax rows
// =====================================================================
__global__ void final_head(const float* __restrict__ h, const float* __restrict__ w,
                           const float* __restrict__ b, float* __restrict__ out) {
  int row = blockIdx.x * blockDim.x + threadIdx.x;
  if (row >= NTOT) return;
  float acc[NCLS];
  #pragma unroll
  for (int j = 0; j < NCLS; ++j) acc[j] = b[j];
  const float* hr = h + (size_t)row * DF;
  for (int k = 0; k < DF; ++k) {
    float v = hr[k];
    #pragma unroll
    for (int j = 0; j < NCLS; ++j) acc[j] += v * w[k * NCLS + j];
  }
  float m = acc[0];
  #pragma unroll
  for (int j = 1; j < NCLS; ++j) m = fmaxf(m, acc[j]);
  float s = 0.f;
  #pragma unroll
  for (int j = 0; j < NCLS; ++j) s += expf(acc[j] - m);
  float l = logf(s);
  #pragma unroll
  for (int j = 0; j < NCLS; ++j) out[(size_t)row * NCLS + j] = acc[j] - m - l;
}

// =====================================================================
extern "C" void kernel_launch(void* const* d_in, const int* in_sizes, int n_in,
                              void* d_out, int out_size, void* d_ws, size_t ws_size,
                              hipStream_t stream) {
  const float* x      = (const float*)d_in[0];
  const int*   ei     = (const int*)d_in[1];          // [2,E] row-major
  const int*   sidx   = (const int*)d_in[2];
  const int*   didx   = (const int*)d_in[3];
  const float* emb_w  = (const float*)d_in[4];
  const float* emb_b  = (const float*)d_in[5];
  const float* gcn1_w = (const float*)d_in[6];
  const float* gcn1_b = (const float*)d_in[7];
  const float* gcn2_w = (const float*)d_in[8];
  const float* gcn2_b = (const float*)d_in[9];
  const float* bn_g   = (const float*)d_in[10];
  const float* bn_b   = (const float*)d_in[11];
  const float* ff_w   = (const float*)d_in[12];       // [384,128]
  const float* ff_b   = (const float*)d_in[13];
  const float* ff1_w  = (const float*)d_in[14];
  const float* ff1_b  = (const float*)d_in[15];
  const float* ff2_w  = (const float*)d_in[16];       // [128,8]
  const float* ff2_b  = (const float*)d_in[17];

  const int* e_src = ei;
  const int* e_dst = ei + EDG;

  // workspace layout (two rotating 32MB activation buffers + small scratch)
  float* big0  = (float*)d_ws;
  float* big1  = big0 + (size_t)NTOT * DF;
  float* dinv  = big1 + (size_t)NTOT * DF;
  float* gbias = dinv + NTOT;
  float* stats = gbias + NB * DF;

  const dim3 blk256(256), blk128(128);
  const int gemm_grid = NTOT / 16;           // 4096

  // 1) symmetric normalization coefficients
  deg_init  <<<NTOT / 256, blk256, 0, stream>>>(dinv);
  deg_count <<<EDG / 256,  blk256, 0, stream>>>(e_dst, dinv);
  deg_finish<<<NTOT / 256, blk256, 0, stream>>>(dinv);

  // 2) embedding: h0 = x @ emb_w + emb_b        (big0)
  gemm_k128_wmma<<<gemm_grid, blk256, 0, stream>>>(x, emb_w, emb_b, nullptr, big0, F_BIAS);

  // 3) GCN1: t1 = h0 @ gcn1_w                   (big1)
  gemm_k128_wmma<<<gemm_grid, blk256, 0, stream>>>(big0, gcn1_w, nullptr, nullptr, big1, 0);
  //    h1(pre-relu) = aggregate(t1) + b         (big0)
  agg_init    <<<NTOT * DF / 4 / 256, blk256, 0, stream>>>(big1, dinv, gcn1_b, big0);
  edge_scatter<<<EDG / 8, blk256, 0, stream>>>(big1, e_src, e_dst, dinv, big0);

  // 4) GCN2: t2 = relu(h1) @ gcn2_w             (big1, relu fused into A-load)
  gemm_k128_wmma<<<gemm_grid, blk256, 0, stream>>>(big0, gcn2_w, nullptr, nullptr, big1, F_RELU_IN);
  //    h2 = aggregate(t2) + b                   (big0)
  agg_init    <<<NTOT * DF / 4 / 256, blk256, 0, stream>>>(big1, dinv, gcn2_b, big0);
  edge_scatter<<<EDG / 8, blk256, 0, stream>>>(big1, e_src, e_dst, dinv, big0);

  // 5) per-graph bias from src/dest vectors (rows 128..383 of ff_w + ff_b)
  graph_bias<<<NB, blk128, 0, stream>>>(big0, sidx, didx, ff_w, ff_b, gbias);
  //    h3 = h2 @ ff_w[0:128] + gbias[graph]     (big1)
  gemm_k128_wmma<<<gemm_grid, blk256, 0, stream>>>(big0, ff_w, nullptr, gbias, big1, F_GBIAS);

  // 6) batchnorm (batch stats) + relu           (big0)
  stats_zero<<<1, blk256, 0, stream>>>(stats);
  bn_reduce <<<NTOT / 128, blk128, 0, stream>>>(big1, stats);
  bn_apply  <<<NTOT * DF / 256, blk256, 0, stream>>>(big1, stats, bn_g, bn_b, big0);

  // 7) h4 = relu(hn @ ff1_w + ff1_b)            (big1)
  gemm_k128_wmma<<<gemm_grid, blk256, 0, stream>>>(big0, ff1_w, ff1_b, nullptr, big1, F_BIAS | F_RELU_OUT);

  // 8) log_softmax(h4 @ ff2_w + ff2_b)          (d_out)
  final_head<<<NTOT / 256, blk256, 0, stream>>>(big1, ff2_w, ff2_b, (float*)d_out);
}